// FeatureFinetuningLoss_14147622273725
// MI455X (gfx1250) — compile-verified
//
#include <hip/hip_runtime.h>

// FeatureFinetuningLoss for MI455X (gfx1250, wave32).
//
// feat, feat_p : (128, 1024, 14, 14) f32   -> 2 x 102.8 MB streamed once
// avg_feat     : (2, 1024) f32 (tiny)
// qual, label  : (128,) int
// out          : scalar f32
//
// Memory-bound (205 MB @ 23.3 TB/s ~ 9 us). The HW-mean is a row-sum done
// with chained V_WMMA_F32_16X16X4_F32 against an all-ones B matrix: one
// b64 load per operand per WMMA step (the f32 A layout matches a natural
// float2 load), so the reduction needs 1 matrix op per 256 bytes/wave.

typedef float v2f __attribute__((ext_vector_type(2)));
typedef float v8f __attribute__((ext_vector_type(8)));

#define HW_N   196          // 14*14
#define C_DIM  1024
#define B_DIM  128
#define MARGIN_V 1.0f
#define EPS_V    1e-6f

__global__ __launch_bounds__(256) void ftl_wmma_kernel(
    const float* __restrict__ feat,
    const float* __restrict__ feat_p,
    const float* __restrict__ avg_feat,
    const int*   __restrict__ qual,
    const int*   __restrict__ label,
    float*       __restrict__ out)
{
    __shared__ float block_acc;
    if (threadIdx.x == 0) block_acc = 0.0f;
    __syncthreads();

    const int lane = threadIdx.x & 31;
    const int wave = threadIdx.x >> 5;                         // 0..7
    const int row0 = (blockIdx.x * 8 + wave) * 16;             // first (b,c) row of tile

    // f32 WMMA A-matrix layout (16x4): lane L holds row M = L&15;
    // lanes 0-15 carry K = {k0, k0+1}, lanes 16-31 carry K = {k0+2, k0+3}.
    const int m     = lane & 15;
    const int khalf = (lane >> 4) * 2;

    const float* fbase  = feat   + (long long)(row0 + m) * HW_N + khalf;
    const float* fpbase = feat_p + (long long)(row0 + m) * HW_N + khalf;

    v8f acc_f  = {};
    v8f acc_fp = {};
    v2f ones;
    ones[0] = 1.0f;
    ones[1] = 1.0f;

    // 196 = 49 * 4 exactly -> no tail. Each step: one b64 load per tensor
    // (naturally 8B aligned: 196*4, k0*4, khalf*4 are all multiples of 8).
    #pragma unroll 7
    for (int k0 = 0; k0 < HW_N; k0 += 4) {
        v2f a  = *(const v2f*)(fbase  + k0);
        v2f ap = *(const v2f*)(fpbase + k0);
        acc_f  = __builtin_amdgcn_wmma_f32_16x16x4_f32(
                     false, a,  false, ones, (short)0, acc_f,  false, false);
        acc_fp = __builtin_amdgcn_wmma_f32_16x16x4_f32(
                     false, ap, false, ones, (short)0, acc_fp, false, false);
    }

    // D layout: lane L<16 -> N=L, VGPR i -> M=i ; lane L>=16 -> N=L-16, M=8+i.
    // B was all-ones so every column equals the row sum; take column N==0
    // (lane 0 -> rows 0..7, lane 16 -> rows 8..15) to count each row once.
    float part = 0.0f;
    if ((lane & 15) == 0) {
        const int   mrow0  = (lane >> 4) * 8;
        const float inv_hw = 1.0f / (float)HW_N;
        const float inv_c  = 1.0f / (float)C_DIM;
        #pragma unroll
        for (int i = 0; i < 8; ++i) {
            const int r = row0 + mrow0 + i;            // global (b,c) row
            const int b = r >> 10;                     // / 1024
            const int c = r & (C_DIM - 1);
            const float mf  = acc_f[i]  * inv_hw;
            const float mfp = acc_fp[i] * inv_hw;
            const float sel = (qual[b] == 1) ? mf : mfp;
            const int   lab = label[b];
            const float pos = avg_feat[lab * C_DIM + c];
            const float neg = avg_feat[(1 - lab) * C_DIM + c];
            const float d_ap = fabsf(sel - pos + EPS_V);
            const float d_an = fabsf(sel - neg + EPS_V);
            const float v = d_ap - d_an + MARGIN_V;
            part += (v > 0.0f ? v : 0.0f) * inv_c;     // mean over C folded in
        }
        atomicAdd(&block_acc, part);                   // ds_add_f32
    }
    __syncthreads();
    if (threadIdx.x == 0) atomicAdd(out, block_acc);   // 1024 global atomics
}

extern "C" void kernel_launch(void* const* d_in, const int* in_sizes, int n_in,
                              void* d_out, int out_size, void* d_ws, size_t ws_size,
                              hipStream_t stream) {
    const float* feat     = (const float*)d_in[0];
    const float* feat_p   = (const float*)d_in[1];
    const float* avg_feat = (const float*)d_in[2];
    const int*   qual     = (const int*)d_in[3];
    const int*   label    = (const int*)d_in[4];
    float* out = (float*)d_out;

    // Zero the scalar accumulator (graph-capture safe).
    hipMemsetAsync(out, 0, sizeof(float), stream);

    // 128*1024 rows / (16 rows per wave * 8 waves per block) = 1024 blocks.
    ftl_wmma_kernel<<<dim3(1024), dim3(256), 0, stream>>>(
        feat, feat_p, avg_feat, qual, label, out);
}